// FrequencySinkhornMatcher_87643102642303
// MI455X (gfx1250) — compile-verified
//
#include <hip/hip_runtime.h>
#include <hip/hip_bf16.h>
#include <math.h>

// ---- problem constants (match reference) ----
#define B_   32
#define Q_   1500
#define QP_  1504          // Q padded to multiple of 16 (pad rows of K are zero)
#define NC_  10
#define T_   400           // multiple of 16 and 4
#define EPS_      1e-6f
#define SINKEPS_  0.1f
#define ALPHA_    0.25f
#define NWDC_     0.05f

typedef float v2f __attribute__((ext_vector_type(2)));
typedef float v8f __attribute__((ext_vector_type(8)));

// ---- monotonic float<->uint key for atomicMax on signed floats ----
__device__ __forceinline__ unsigned fenc(float f) {
    unsigned u = __float_as_uint(f);
    return (u & 0x80000000u) ? ~u : (u | 0x80000000u);
}
__device__ __forceinline__ float fdec(unsigned k) {
    unsigned u = (k & 0x80000000u) ? (k & 0x7fffffffu) : ~k;
    return __uint_as_float(u);
}

// ---- kernel 0: init v = 1/T, maxbuf = 0 ----
__global__ void fsm_init(float* __restrict__ vbuf, unsigned* __restrict__ maxbuf) {
    int i = blockIdx.x * blockDim.x + threadIdx.x;
    if (i < B_ * T_) vbuf[i] = 1.0f / (float)T_;
    if (i < B_) maxbuf[i] = 0u;
}

// ---- kernel 1: focal class-cost table ccls[b][q][c] (only B*Q*NC distinct values) ----
__global__ void fsm_classcost(const float* __restrict__ logits,
                              float* __restrict__ ccls) {
    int i = blockIdx.x * blockDim.x + threadIdx.x;   // over B*Q*NC
    if (i >= B_ * Q_ * NC_) return;
    float x = logits[i];
    float p = 1.0f / (1.0f + expf(-x));
    float omp = 1.0f - p;
    float neg = (1.0f - ALPHA_) * p * p * (-logf(omp + 1e-8f));
    float pos = ALPHA_ * omp * omp * (-logf(p + 1e-8f));
    ccls[i] = pos - neg;                             // cost_class = pos - neg
}

// ---- kernel 2: C = 5*bbox + class + 5*nwd, stored into Kmat; per-batch max via atomicMax ----
__global__ void fsm_cost_max(const float* __restrict__ ccls,
                             const float* __restrict__ pboxes,
                             const int*   __restrict__ labels,
                             const float* __restrict__ tboxes,
                             float* __restrict__ Kmat,
                             unsigned* __restrict__ maxbuf) {
    __shared__ float smax[256];
    const int b = blockIdx.y;
    const int tid = threadIdx.x;
    long idx = (long)blockIdx.x * blockDim.x + tid;
    float c = -3.0e38f;
    if (idx < (long)Q_ * T_) {
        int q = (int)(idx / T_);
        int t = (int)(idx % T_);
        int lab = labels[b * T_ + t];
        float cost_class = ccls[((size_t)b * Q_ + q) * NC_ + lab];
        float4 pb = *(const float4*)(pboxes + ((size_t)b * Q_ + q) * 4);  // 16B aligned
        float4 tb = *(const float4*)(tboxes + ((size_t)b * T_ + t) * 4);  // 16B aligned
        float d0 = pb.x - tb.x, d1 = pb.y - tb.y, d2 = pb.z - tb.z, d3 = pb.w - tb.w;
        float cost_bbox = fabsf(d0) + fabsf(d1) + fabsf(d2) + fabsf(d3);
        float w2 = d0 * d0 + d1 * d1 + 0.25f * d2 * d2 + 0.25f * d3 * d3;
        float nwd = expf(-sqrtf(fmaxf(w2, 1e-7f)) / NWDC_);
        c = 5.0f * cost_bbox + cost_class + 5.0f * (1.0f - nwd);
        Kmat[(size_t)b * QP_ * T_ + idx] = c;        // stash C; exp'd in next pass (L2-resident)
    }
    smax[tid] = c;
    __syncthreads();
    for (int s = 128; s > 0; s >>= 1) {
        if (tid < s) smax[tid] = fmaxf(smax[tid], smax[tid + s]);
        __syncthreads();
    }
    if (tid == 0) atomicMax(maxbuf + b, fenc(smax[0]));
}

// ---- kernel 3: K = exp(-(C - Cmax)/eps) in place; zero pad rows ----
__global__ void fsm_kbuild(const unsigned* __restrict__ maxbuf,
                           float* __restrict__ Kmat) {
    const int b = blockIdx.y;
    long idx = (long)blockIdx.x * blockDim.x + threadIdx.x;
    if (idx >= (long)QP_ * T_) return;
    int q = (int)(idx / T_);
    float* dst = Kmat + (size_t)b * QP_ * T_ + idx;
    if (q >= Q_) { *dst = 0.0f; return; }
    float cmax = fdec(maxbuf[b]);
    *dst = expf(-(*dst - cmax) / SINKEPS_);
}

// ---- kernel 4: u = 1/(K v + eps) for one 16-row chunk, via V_WMMA_F32_16X16X4_F32 ----
// A = 16x4 tile of K (lane 16h+m holds K[row0+m][k+2h .. k+2h+1], one b64 load).
// B = v[k..k+3] broadcast across the 16 columns -> all D columns identical.
__global__ void fsm_sink_u(const float* __restrict__ Kmat,
                           const float* __restrict__ vbuf,
                           float* __restrict__ ubuf) {
    const int chunk = blockIdx.x;            // 0 .. QP_/16-1
    const int b = blockIdx.y;
    const int lane = threadIdx.x;            // 0..31 (wave32)
    const int half = lane >> 4;              // 0 or 1
    const int m = lane & 15;
    const int row0 = chunk * 16;
    const float* Kb = Kmat + (size_t)b * QP_ * T_;
    const float* vb = vbuf + b * T_;
    v8f acc = {};
    for (int k = 0; k < T_; k += 4) {
        const float* ap = Kb + (size_t)(row0 + m) * T_ + k + 2 * half;
        v2f a = *(const v2f*)ap;                              // 8B aligned
        v2f bb = *(const v2f*)(vb + k + 2 * half);            // 8B aligned broadcast pair
        acc = __builtin_amdgcn_wmma_f32_16x16x4_f32(
            false, a, false, bb, (short)0, acc, false, false);
    }
    // D columns identical: lane 0 holds rows 0..7, lane 16 holds rows 8..15.
    float* ub = ubuf + b * QP_ + row0;
    if (lane == 0) {
#pragma unroll
        for (int j = 0; j < 8; ++j) ub[j] = 1.0f / (acc[j] + EPS_);
    } else if (lane == 16) {
#pragma unroll
        for (int j = 0; j < 8; ++j) ub[8 + j] = 1.0f / (acc[j] + EPS_);
    }
}

// ---- kernel 5: v = 1/(K^T u + eps) for one 16-col chunk ----
// A = u[q..q+3] broadcast across the 16 rows; B = 4x16 tile of K (lane n holds column t0+n,
// contiguous across lanes -> coalesced). All D rows identical: lane n's acc[0] = result t0+n.
__global__ void fsm_sink_v(const float* __restrict__ Kmat,
                           const float* __restrict__ ubuf,
                           float* __restrict__ vbuf) {
    const int chunk = blockIdx.x;            // 0 .. T_/16-1
    const int b = blockIdx.y;
    const int lane = threadIdx.x;
    const int half = lane >> 4;
    const int n = lane & 15;
    const int t0 = chunk * 16;
    const float* Kb = Kmat + (size_t)b * QP_ * T_;
    const float* ub = ubuf + b * QP_;
    v8f acc = {};
    for (int q = 0; q < QP_; q += 4) {
        int qq = q + 2 * half;
        v2f a = *(const v2f*)(ub + qq);                       // 8B aligned broadcast pair
        v2f bb;
        bb[0] = Kb[(size_t)qq * T_ + t0 + n];
        bb[1] = Kb[(size_t)(qq + 1) * T_ + t0 + n];
        acc = __builtin_amdgcn_wmma_f32_16x16x4_f32(
            false, a, false, bb, (short)0, acc, false, false);
    }
    if (lane < 16) vbuf[b * T_ + t0 + lane] = 1.0f / (acc[0] + EPS_);
}

// ---- kernel 6: src_idx[b][t] = argmax_q u[q]*K[q][t]; tgt_idx[b][t] = t ----
// (v[t] > 0 is constant per column, so it drops out of the argmax.)
__global__ void fsm_argmax(const float* __restrict__ Kmat,
                           const float* __restrict__ ubuf,
                           float* __restrict__ out) {
    const int b = blockIdx.y;
    int t = blockIdx.x * blockDim.x + threadIdx.x;
    if (t >= T_) return;
    const float* Kb = Kmat + (size_t)b * QP_ * T_ + t;
    const float* ub = ubuf + b * QP_;
    float best = -1.0f;
    int bi = 0;
    for (int q = 0; q < Q_; ++q) {
        float val = ub[q] * Kb[(size_t)q * T_];   // coalesced across t for each q
        if (val > best) { best = val; bi = q; }   // keep first max (jnp.argmax tie rule)
    }
    out[b * T_ + t] = (float)bi;                      // src_idx
    out[(size_t)B_ * T_ + b * T_ + t] = (float)t;     // tgt_idx
}

extern "C" void kernel_launch(void* const* d_in, const int* in_sizes, int n_in,
                              void* d_out, int out_size, void* d_ws, size_t ws_size,
                              hipStream_t stream) {
    const float* logits = (const float*)d_in[0];   // (B,Q,NC)
    const float* pboxes = (const float*)d_in[1];   // (B,Q,4)
    const int*   labels = (const int*)d_in[2];     // (B,T)
    const float* tboxes = (const float*)d_in[3];   // (B,T,4)
    float* out = (float*)d_out;                    // src_idx (B*T) ++ tgt_idx (B*T)

    // workspace layout (~79 MB total)
    char* ws = (char*)d_ws;
    float* Kmat = (float*)ws;                                    // B*QP*T floats (~77MB)
    float* ubuf = (float*)(ws + (size_t)B_ * QP_ * T_ * 4);      // B*QP floats
    float* vbuf = ubuf + (size_t)B_ * QP_;                       // B*T floats
    unsigned* maxbuf = (unsigned*)(vbuf + (size_t)B_ * T_);      // B uints
    float* ccls = (float*)(maxbuf + B_);                         // B*Q*NC floats (~1.9MB)

    // 0) init v = 1/T, maxbuf = 0
    fsm_init<<<(B_ * T_ + 255) / 256, 256, 0, stream>>>(vbuf, maxbuf);

    // 1) focal class-cost table (B*Q*NC distinct values instead of B*Q*T)
    fsm_classcost<<<(B_ * Q_ * NC_ + 255) / 256, 256, 0, stream>>>(logits, ccls);

    // 2) cost matrix -> C stored in Kmat, per-batch max
    {
        dim3 grid((unsigned)(((long)Q_ * T_ + 255) / 256), B_);
        fsm_cost_max<<<grid, 256, 0, stream>>>(ccls, pboxes, labels, tboxes, Kmat, maxbuf);
    }
    // 3) K = exp(-(C - Cmax)/eps) in place (C read back from L2)
    {
        dim3 grid((unsigned)(((long)QP_ * T_ + 255) / 256), B_);
        fsm_kbuild<<<grid, 256, 0, stream>>>(maxbuf, Kmat);
    }
    // 4) 20 Sinkhorn iterations (K lives in L2 across all sweeps)
    for (int it = 0; it < 20; ++it) {
        fsm_sink_u<<<dim3(QP_ / 16, B_), 32, 0, stream>>>(Kmat, vbuf, ubuf);
        fsm_sink_v<<<dim3(T_ / 16, B_), 32, 0, stream>>>(Kmat, ubuf, vbuf);
    }
    // 5) argmax over queries per target column
    fsm_argmax<<<dim3((T_ + 255) / 256, B_), 256, 0, stream>>>(Kmat, ubuf, out);
}